// ChainMessagePassing_71614284693759
// MI455X (gfx1250) — compile-verified
//
#include <hip/hip_runtime.h>
#include <stdint.h>

// ChainMessagePassing: out[dst] += x[src] over two random edge lists.
// Memory-bound scatter-add; see analysis. Strategy:
//  - 16 lanes per edge, float4 per lane -> one coalesced 256B gather per edge
//  - non-temporal (NT temporal hint) loads for the 102MB int64 index stream
//    so it does not thrash the L2-resident x (25.6MB) + out (25.6MB)
//  - agent-scope hardware global_atomic_add_f32 (no CAS loop)
//  - global_prefetch_b8 run-ahead on the index stream

#define THREADS 256
#define PREFETCH_EDGES 2048  // ~16KB ahead on each index row

__global__ __launch_bounds__(THREADS) void scatter_add_edges(
    const float* __restrict__ x,          // [N, 64]
    const long long* __restrict__ eidx,   // [2, E] row0 = src, row1 = dst
    float* __restrict__ out,              // [N, 64]
    long long n_edges)
{
    const long long tid  = (long long)blockIdx.x * THREADS + threadIdx.x;
    const long long edge = tid >> 4;          // 16 lanes per edge
    const int       chunk = (int)(tid & 15);  // which float4 of the 64 feats
    if (edge >= n_edges) return;

    // Run-ahead prefetch of the streaming index rows (global_prefetch_b8).
    if (chunk == 0) {
        long long pe = edge + PREFETCH_EDGES;
        if (pe < n_edges) {
            __builtin_prefetch(&eidx[pe], 0, 0);
            __builtin_prefetch(&eidx[n_edges + pe], 0, 0);
        }
    }

    // Streamed once -> non-temporal so L2 keeps x/out resident.
    const long long src = __builtin_nontemporal_load(&eidx[edge]);
    const long long dst = __builtin_nontemporal_load(&eidx[n_edges + edge]);

    // Coalesced 256B gather: 16 lanes x b128, x stays hot in L2.
    const float4 v = *(const float4*)(x + (src << 6) + ((long long)chunk << 2));

    // Scatter-add: 4 hardware f32 atomics at L2 (agent scope, relaxed).
    float* o = out + (dst << 6) + ((long long)chunk << 2);
    __hip_atomic_fetch_add(o + 0, v.x, __ATOMIC_RELAXED, __HIP_MEMORY_SCOPE_AGENT);
    __hip_atomic_fetch_add(o + 1, v.y, __ATOMIC_RELAXED, __HIP_MEMORY_SCOPE_AGENT);
    __hip_atomic_fetch_add(o + 2, v.z, __ATOMIC_RELAXED, __HIP_MEMORY_SCOPE_AGENT);
    __hip_atomic_fetch_add(o + 3, v.w, __ATOMIC_RELAXED, __HIP_MEMORY_SCOPE_AGENT);
}

extern "C" void kernel_launch(void* const* d_in, const int* in_sizes, int n_in,
                              void* d_out, int out_size, void* d_ws, size_t ws_size,
                              hipStream_t stream) {
    const float*     x        = (const float*)d_in[0];       // [N, 64] fp32
    const long long* up_idx   = (const long long*)d_in[1];   // [2, E] int64
    const long long* down_idx = (const long long*)d_in[2];   // [2, E] int64
    float*           out      = (float*)d_out;               // [N, 64] fp32

    const long long n_edges_up   = (long long)in_sizes[1] / 2;
    const long long n_edges_down = (long long)in_sizes[2] / 2;

    // Zero the accumulator (graph-capturable async memset).
    hipMemsetAsync(d_out, 0, (size_t)out_size * sizeof(float), stream);

    // up pass
    {
        long long total = n_edges_up * 16;
        long long blocks = (total + THREADS - 1) / THREADS;
        scatter_add_edges<<<(uint32_t)blocks, THREADS, 0, stream>>>(
            x, up_idx, out, n_edges_up);
    }
    // down pass (accumulates into the same buffer -> up_out + down_out)
    {
        long long total = n_edges_down * 16;
        long long blocks = (total + THREADS - 1) / THREADS;
        scatter_add_edges<<<(uint32_t)blocks, THREADS, 0, stream>>>(
            x, down_idx, out, n_edges_down);
    }
}